// Decoder_74122545594383
// MI455X (gfx1250) — compile-verified
//
#include <hip/hip_runtime.h>
#include <stdint.h>
#include <stddef.h>

// Problem constants (match reference)
#define B_        8
#define N_        4096
#define C_        91        // TOTAL_LABELS
#define K_        100       // MAX_TOTAL
#define THREADS   256
#define NWAVES    (THREADS / 32)
#define ELEMS     (N_ / THREADS)   // 16 boxes per thread, register-resident
#define SCORE_THR 0.05f
#define IOU_THR   0.5f

typedef unsigned int v4u __attribute__((ext_vector_type(4)));
typedef int          v8i __attribute__((ext_vector_type(8)));
typedef int          v4i __attribute__((ext_vector_type(4)));

// TDM availability (device pass only; ROCm 7.2 = 5-arg builtin)
#if defined(__HIP_DEVICE_COMPILE__) && defined(__gfx1250__) && __has_builtin(__builtin_amdgcn_tensor_load_to_lds)
#define HAVE_TDM 1
#else
#define HAVE_TDM 0
#endif

__device__ __forceinline__ float clamp01(float v) { return fminf(fmaxf(v, 0.0f), 1.0f); }

// argmax merge with jnp.argmax / lax.top_k tie-break (first/lowest index wins)
__device__ __forceinline__ void argmerge(float v, int i, float& bv, int& bi) {
  if (v > bv || (v == bv && i < bi)) { bv = v; bi = i; }
}

// Block-wide argmax: wave32 shuffle reduction (ds path) + tiny LDS merge.
// On return every thread holds the block-best (bv, bi).
// Caller must __syncthreads() before rv/ri may be reused.
__device__ __forceinline__ void block_argmax(float& bv, int& bi, float* rv, int* ri) {
  #pragma unroll
  for (int off = 16; off > 0; off >>= 1) {
    float ov = __shfl_xor(bv, off, 32);
    int   oi = __shfl_xor(bi, off, 32);
    argmerge(ov, oi, bv, bi);
  }
  const int lane = threadIdx.x & 31;
  const int wave = threadIdx.x >> 5;
  if (lane == 0) { rv[wave] = bv; ri[wave] = bi; }
  __syncthreads();
  bv = rv[0]; bi = ri[0];
  #pragma unroll
  for (int w = 1; w < NWAVES; ++w) argmerge(rv[w], ri[w], bv, bi);
}

#if HAVE_TDM
// TDM: DMA a contiguous row of `nelem` f32 from global into LDS (1 tile, 2D degenerate).
// D# layout per CDNA5 ISA ch.8 (group0: count/lds/global/type=2; group1: data_size=4B,
// tensor_dim0 = tile_dim0 = nelem, dim1 = 1, stride0 = nelem). Issue from one wave only.
__device__ __forceinline__ void tdm_load_row_to_lds(const float* gsrc, float* lds_dst, unsigned ne) {
  uint64_t ga = (uint64_t)(uintptr_t)gsrc;
  uint32_t la = (uint32_t)(uintptr_t)lds_dst;  // low 32 bits of flat LDS addr = LDS byte offset
  v4u g0;
  g0[0] = 1u;                                           // count=1, is_restore=0, gather off
  g0[1] = la;                                           // lds_addr
  g0[2] = (uint32_t)ga;                                 // global_addr[31:0]
  g0[3] = (uint32_t)((ga >> 32) & 0x01FFFFFFull) | (2u << 30);  // global_addr[56:32] | type=2
  v8i g1;
  g1[0] = (int)(2u << 16);                              // workgroup_mask=0, data_size=4B, no flags
  g1[1] = (int)((ne & 0xFFFFu) << 16);                  // tensor_dim0[15:0]
  g1[2] = (int)((ne >> 16) | (1u << 16));               // tensor_dim0[31:16] | tensor_dim1=1
  g1[3] = (int)((ne & 0xFFFFu) << 16);                  // tensor_dim1[31:16]=0 | tile_dim0=ne
  g1[4] = 1;                                            // tile_dim1=1, tile_dim2=0
  g1[5] = (int)ne;                                      // tensor_dim0_stride[31:0]
  g1[6] = (int)((ne & 0xFFFFu) << 16);                  // stride0[47:32]=0 | tensor_dim1_stride[15:0]
  g1[7] = (int)(ne >> 16);                              // tensor_dim1_stride[47:16]
  v4i z4 = {0, 0, 0, 0};
#if __has_include(<hip/amd_detail/amd_gfx1250_TDM.h>)
  v8i z8 = {0, 0, 0, 0, 0, 0, 0, 0};
  __builtin_amdgcn_tensor_load_to_lds(g0, g1, z4, z4, z8, 0);   // therock 6-arg form
#else
  __builtin_amdgcn_tensor_load_to_lds(g0, g1, z4, z4, 0);       // ROCm 7.2 5-arg form
#endif
#if __has_builtin(__builtin_amdgcn_s_wait_tensorcnt)
  __builtin_amdgcn_s_wait_tensorcnt(0);
#else
  asm volatile("s_wait_tensorcnt 0" ::: "memory");
#endif
}
#endif

// ---------------------------------------------------------------------------
// Kernel 1: per-anchor argmax over 91 classes -> background mask (argmax != 0)
// ---------------------------------------------------------------------------
__global__ void __launch_bounds__(THREADS)
mask_kernel(const float* __restrict__ probs, float* __restrict__ mask) {
  const int i = blockIdx.x * THREADS + threadIdx.x;   // i in [0, B*N)
  const float* p = probs + (size_t)i * C_;
  float mv = p[0];
  int   mc = 0;
  #pragma unroll 7
  for (int c = 1; c < C_; ++c) {
    float v = p[c];
    if (v > mv) { mv = v; mc = c; }   // strict > == first-index tie-break
  }
  mask[i] = (mc != 0) ? 1.0f : 0.0f;
}

// ---------------------------------------------------------------------------
// Kernel 2: one block per (b, c): decode 4096 class-c boxes into registers,
// greedy NMS with register-resident live scores, early exit, zero-fill tail.
// ---------------------------------------------------------------------------
__global__ void __launch_bounds__(THREADS)
nms_kernel(const float* __restrict__ roi, const float* __restrict__ deltas,
           const float* __restrict__ probs, const float* __restrict__ mask,
           float* __restrict__ sel_s, float* __restrict__ sel_b) {
  const int bc  = blockIdx.x;
  const int b   = bc / C_;
  const int c   = bc % C_;
  const int tid = threadIdx.x;
  const int bN  = b * N_;

  __shared__ float rv[NWAVES];
  __shared__ int   ri[NWAVES];
  __shared__ float sbox[4];

  float y1[ELEMS], x1[ELEMS], y2[ELEMS], x2[ELEMS], live[ELEMS];

  const float4* roi4 = (const float4*)roi;
  const float4* dlt4 = (const float4*)deltas;

  // Warm the L2/WGP$ path for the 1456B-strided delta gather (global_prefetch_b8)
  #pragma unroll
  for (int i = 0; i < ELEMS; ++i) {
    const int n = tid + (i << 8);
    __builtin_prefetch((const void*)&dlt4[(size_t)(bN + n) * C_ + c], 0, 1);
  }

  // Decode: [dy,dx,dh,dw] * [0.1,0.1,0.2,0.2], center-form, clip to [0,1]
  #pragma unroll
  for (int i = 0; i < ELEMS; ++i) {
    const int n = tid + (i << 8);
    const float4 r = roi4[bN + n];                          // y1 x1 y2 x2
    const float4 d = dlt4[(size_t)(bN + n) * C_ + c];
    const float ah = r.z - r.x, aw = r.w - r.y;
    const float acy = r.x + 0.5f * ah, acx = r.y + 0.5f * aw;
    const float h  = expf(d.z * 0.2f) * ah;
    const float w  = expf(d.w * 0.2f) * aw;
    const float cy = d.x * 0.1f * ah + acy;
    const float cx = d.y * 0.1f * aw + acx;
    y1[i] = clamp01(cy - 0.5f * h);
    x1[i] = clamp01(cx - 0.5f * w);
    y2[i] = clamp01(cy + 0.5f * h);
    x2[i] = clamp01(cx + 0.5f * w);
    const float p = probs[(size_t)(bN + n) * C_ + c];
    const float s = (mask[bN + n] != 0.0f) ? p : 0.0f;      // zero scores of bg anchors
    live[i] = (s > SCORE_THR) ? s : -1.0f;
  }

  const int obase = bc * K_;
  int k = 0;
  for (; k < K_; ++k) {
    // global argmax over live scores (first-index tie-break)
    float bv = -2.0f; int bi = 0x7fffffff;
    #pragma unroll
    for (int i = 0; i < ELEMS; ++i) argmerge(live[i], tid + (i << 8), bv, bi);
    block_argmax(bv, bi, rv, ri);

    if (bv <= 0.0f) break;                                  // no live boxes remain

    // owner thread publishes the winning box (static register indices only)
    #pragma unroll
    for (int i = 0; i < ELEMS; ++i) {
      if (tid + (i << 8) == bi) { sbox[0] = y1[i]; sbox[1] = x1[i]; sbox[2] = y2[i]; sbox[3] = x2[i]; }
    }
    __syncthreads();
    const float b0 = sbox[0], b1 = sbox[1], b2 = sbox[2], b3 = sbox[3];
    const float areaB = (b2 - b0) * (b3 - b1);

    // suppress: iou > 0.5 (division, 1e-8 clamp, like reference) + explicit self-kill
    #pragma unroll
    for (int i = 0; i < ELEMS; ++i) {
      const float iy1 = fmaxf(b0, y1[i]), ix1 = fmaxf(b1, x1[i]);
      const float iy2 = fminf(b2, y2[i]), ix2 = fminf(b3, x2[i]);
      const float inter = fmaxf(iy2 - iy1, 0.0f) * fmaxf(ix2 - ix1, 0.0f);
      const float areaN = (y2[i] - y1[i]) * (x2[i] - x1[i]);
      const float u = fmaxf(areaB + areaN - inter, 1e-8f);
      if ((inter / u) > IOU_THR || (tid + (i << 8)) == bi) live[i] = -1.0f;
    }

    if (tid == 0) {
      sel_s[obase + k] = bv;
      float4 bb; bb.x = b0; bb.y = b1; bb.z = b2; bb.w = b3;
      ((float4*)sel_b)[obase + k] = bb;
    }
    __syncthreads();   // protects sbox / rv / ri for next round
  }

  // zero-fill the tail (d_ws is poisoned; invalid picks must be exact zeros)
  for (int j = k + tid; j < K_; j += THREADS) {
    sel_s[obase + j] = 0.0f;
    float4 z; z.x = z.y = z.z = z.w = 0.0f;
    ((float4*)sel_b)[obase + j] = z;
  }
}

// ---------------------------------------------------------------------------
// Kernel 3: one block per batch: top-100 of 9100 candidates (TDM -> LDS),
// emit final_bboxes / final_labels / final_scores.
// ---------------------------------------------------------------------------
__global__ void __launch_bounds__(THREADS)
topk_kernel(const float* __restrict__ sel_s, const float* __restrict__ sel_b,
            float* __restrict__ out) {
  const int b   = blockIdx.x;
  const int tid = threadIdx.x;

  __shared__ float sc[C_ * K_];          // 9100 floats = 36.4 KB
  __shared__ float rv[NWAVES];
  __shared__ int   ri[NWAVES];

#if HAVE_TDM
  if (tid < 32) {                        // one wave issues the DMA (EXEC ignored by TDM)
    tdm_load_row_to_lds(sel_s + (size_t)b * (C_ * K_), sc, (unsigned)(C_ * K_));
  }
#else
  for (int j = tid; j < C_ * K_; j += THREADS)
    sc[j] = sel_s[(size_t)b * (C_ * K_) + j];
#endif
  __syncthreads();

  float* out_boxes  = out;                         // [B,100,4]
  float* out_labels = out + B_ * K_ * 4;           // [B,100]
  float* out_scores = out + B_ * K_ * 4 + B_ * K_; // [B,100]

  for (int k = 0; k < K_; ++k) {
    float bv = -2.0f; int bi = 0x7fffffff;
    for (int j = tid; j < C_ * K_; j += THREADS) argmerge(sc[j], j, bv, bi);
    block_argmax(bv, bi, rv, ri);

    if (tid == 0) {
      sc[bi] = -1.0f;                                        // remove (below any score >= 0)
      out_scores[b * K_ + k] = bv;
      out_labels[b * K_ + k] = (bv > 0.0f) ? (float)(bi / K_) : 0.0f;  // class id; 0 if invalid
      const float4 bb = ((const float4*)sel_b)[(size_t)b * (C_ * K_) + bi];
      ((float4*)out_boxes)[b * K_ + k] = bb;
    }
    __syncthreads();
  }
}

// ---------------------------------------------------------------------------
extern "C" void kernel_launch(void* const* d_in, const int* in_sizes, int n_in,
                              void* d_out, int out_size, void* d_ws, size_t ws_size,
                              hipStream_t stream) {
  (void)in_sizes; (void)n_in; (void)out_size; (void)ws_size;
  const float* roi    = (const float*)d_in[0];   // [8,4096,4]
  const float* deltas = (const float*)d_in[1];   // [8,4096,364]
  const float* probs  = (const float*)d_in[2];   // [8,4096,91]
  float* out = (float*)d_out;                    // 3200 + 800 + 800 f32

  float* ws    = (float*)d_ws;
  float* mask  = ws;                             // B*N                = 32768 f32
  float* sel_s = ws + (size_t)B_ * N_;           // B*C*K              = 72800 f32
  float* sel_b = sel_s + (size_t)B_ * C_ * K_;   // B*C*K*4 (16B-aligned offset)

  mask_kernel<<<(B_ * N_) / THREADS, THREADS, 0, stream>>>(probs, mask);
  nms_kernel <<<B_ * C_,            THREADS, 0, stream>>>(roi, deltas, probs, mask, sel_s, sel_b);
  topk_kernel<<<B_,                 THREADS, 0, stream>>>(sel_s, sel_b, out);
}